// model_79422535238385
// MI455X (gfx1250) — compile-verified
//
#include <hip/hip_runtime.h>
#include <hip/hip_bf16.h>

// CDNA5 / gfx1250: wave32, WMMA 16x16x4 f32, fragment-ready LDS weights,
// one-deep software-pipelined K loops.
typedef float v2f __attribute__((ext_vector_type(2)));
typedef float v8f __attribute__((ext_vector_type(8)));

#define NN 16384
#define DD 16
#define PP 8
#define MM 50
#define HH 128

// k-pair weight layout: element (pair p, col c, k&1) at  p*STRIDE + c*2 + (k&1)
// -> a lane's B fragment {W[k0][col], W[k0+1][col]} is one aligned ds_load_b64.
// STRIDE % 64 == 32 so the two 16-lane halves (pair p vs p+1) hit disjoint banks.
#define PW12 288   // pair stride for W1 (12 pairs) and W2 (64 pairs)
#define PW3   32   // pair stride for W3 (64 pairs, 16 cols)
#define SXY   28   // per-wave [16][24] X|Y scratch stride; 28r mod 64 distinct for r=0..15
#define SH   132   // per-wave [16][128] activation scratch; 132r mod 64 = 4r -> conflict-free b64
#define WAVES 4    // waves per block, 16 rows each -> 64 rows/block

static __device__ __forceinline__ v2f ldsv2(const float* p) { return *(const v2f*)p; }

__launch_bounds__(128, 2)
__global__ void sde_rollout_kernel(const float* __restrict__ X0,
                                   const float* __restrict__ V0,
                                   const float* __restrict__ Yobs,
                                   const float* __restrict__ noise,
                                   const float* __restrict__ W1,
                                   const float* __restrict__ b1,
                                   const float* __restrict__ W2,
                                   const float* __restrict__ b2,
                                   const float* __restrict__ W3,
                                   const float* __restrict__ b3,
                                   float* __restrict__ outX,
                                   float* __restrict__ outV)
{
    __shared__ float sW1p[12 * PW12];   //  3456 dw : W1 rows 1..24 in k-pair layout
    __shared__ float sW1t[HH];          //   128 dw : t-row of W1
    __shared__ float sB1 [HH];
    __shared__ float sW2p[64 * PW12];   // 18432 dw : W2 in k-pair layout
    __shared__ float sB2 [HH];
    __shared__ float sW3p[64 * PW3];    //  2048 dw : W3 in k-pair layout
    __shared__ float sB3 [DD];
    __shared__ float sXY [WAVES][16 * SXY];  // per-wave [X | Y] input scratch
    __shared__ float sHH [WAVES][16 * SH];   // per-wave activation scratch (h1 then h2)

    const int tid  = threadIdx.x;
    const int lane = tid & 31;
    const int wave = tid >> 5;
    const int lo   = lane & 15;   // column index inside tile
    const int hi   = lane >> 4;   // half-wave selector
    const int rowbase = blockIdx.x * (WAVES * 16) + wave * 16;

    // ---------------- cooperative weight staging (once) ----------------
    for (int i = tid; i < 24 * HH; i += blockDim.x) {
        int r = i >> 7, c = i & 127;
        sW1p[(r >> 1) * PW12 + c * 2 + (r & 1)] = W1[(1 + r) * HH + c];  // rows 1..24
    }
    if (tid < HH) { sW1t[tid] = W1[tid]; sB1[tid] = b1[tid]; sB2[tid] = b2[tid]; }
    for (int i = tid; i < HH * HH; i += blockDim.x) {
        int r = i >> 7, c = i & 127;
        sW2p[(r >> 1) * PW12 + c * 2 + (r & 1)] = W2[i];
    }
    for (int i = tid; i < HH * DD; i += blockDim.x) {
        int r = i >> 4, c = i & 15;
        sW3p[(r >> 1) * PW3 + c * 2 + (r & 1)] = W3[i];
    }
    if (tid < DD) sB3[tid] = b3[tid];

    // stage static observations into per-wave scratch, cols 16..23
    float* xy = sXY[wave];
    for (int i = lane; i < 16 * PP; i += 32) {
        int r = i >> 3, c = i & 7;
        xy[r * SXY + DD + c] = Yobs[(rowbase + r) * PP + c];
    }
    __syncthreads();

    // ---------------- per-wave state in C/D tile layout -----------------
    float xr[8], va[8];
    #pragma unroll
    for (int v = 0; v < 8; ++v) {
        int row = rowbase + v + 8 * hi;
        xr[v] = X0[row * DD + lo];
        va[v] = V0[row];
    }

    const float dt    = 1.0f / MM;
    const float sqdt  = __builtin_sqrtf(dt);
    const float decay = 1.0f - dt;
    float* hsc = sHH[wave];
    const v8f zc = {0.f, 0.f, 0.f, 0.f, 0.f, 0.f, 0.f, 0.f};

    for (int m = 0; m < MM; ++m) {
        const float t = m * dt;

        // stage X_m into [X|Y] scratch (D-layout -> row-major)
        #pragma unroll
        for (int v = 0; v < 8; ++v)
            xy[(v + 8 * hi) * SXY + lo] = xr[v];

        // Brownian tile for this step, same layout as xr / Z
        float ep[8];
        #pragma unroll
        for (int v = 0; v < 8; ++v) {
            size_t row = (size_t)m * NN + (size_t)(rowbase + v + 8 * hi);
            ep[v] = noise[row * DD + lo];
        }

        asm volatile("s_wait_dscnt 0" ::: "memory");  // X stores visible to transposed reads

        v8f acc[8];

        // ========== layer 1: h1 = relu([t,X,Y] @ W1 + b1), K = 24 ==========
        {
            v2f a = ldsv2(&xy[lo * SXY + 2 * hi]);
            v2f b[8];
            #pragma unroll
            for (int c = 0; c < 8; ++c) b[c] = ldsv2(&sW1p[hi * PW12 + (c * 16 + lo) * 2]);
            #pragma unroll
            for (int kk = 0; kk < 6; ++kk) {
                v2f an = a; v2f bn[8];
                #pragma unroll
                for (int c = 0; c < 8; ++c) bn[c] = b[c];
                if (kk < 5) {               // prefetch next k-step before issuing WMMAs
                    an = ldsv2(&xy[lo * SXY + 4 * (kk + 1) + 2 * hi]);
                    #pragma unroll
                    for (int c = 0; c < 8; ++c)
                        bn[c] = ldsv2(&sW1p[(2 * (kk + 1) + hi) * PW12 + (c * 16 + lo) * 2]);
                }
                #pragma unroll
                for (int c = 0; c < 8; ++c)
                    acc[c] = __builtin_amdgcn_wmma_f32_16x16x4_f32(
                        false, a, false, b[c], (short)0, (kk == 0 ? zc : acc[c]), false, false);
                a = an;
                #pragma unroll
                for (int c = 0; c < 8; ++c) b[c] = bn[c];
            }
        }
        #pragma unroll
        for (int c = 0; c < 8; ++c) {
            float bias = t * sW1t[c * 16 + lo] + sB1[c * 16 + lo];  // t-column + b1
            #pragma unroll
            for (int v = 0; v < 8; ++v)
                hsc[(v + 8 * hi) * SH + c * 16 + lo] = fmaxf(acc[c][v] + bias, 0.0f);
        }
        asm volatile("s_wait_dscnt 0" ::: "memory");

        // ========== layer 2: h2 = relu(h1 @ W2 + b2), K = 128 ==========
        {
            v2f a = ldsv2(&hsc[lo * SH + 2 * hi]);
            v2f b[8];
            #pragma unroll
            for (int c = 0; c < 8; ++c) b[c] = ldsv2(&sW2p[hi * PW12 + (c * 16 + lo) * 2]);
            #pragma unroll
            for (int kk = 0; kk < 32; ++kk) {
                v2f an = a; v2f bn[8];
                #pragma unroll
                for (int c = 0; c < 8; ++c) bn[c] = b[c];
                if (kk < 31) {
                    an = ldsv2(&hsc[lo * SH + 4 * (kk + 1) + 2 * hi]);
                    #pragma unroll
                    for (int c = 0; c < 8; ++c)
                        bn[c] = ldsv2(&sW2p[(2 * (kk + 1) + hi) * PW12 + (c * 16 + lo) * 2]);
                }
                #pragma unroll
                for (int c = 0; c < 8; ++c)
                    acc[c] = __builtin_amdgcn_wmma_f32_16x16x4_f32(
                        false, a, false, b[c], (short)0, (kk == 0 ? zc : acc[c]), false, false);
                a = an;
                #pragma unroll
                for (int c = 0; c < 8; ++c) b[c] = bn[c];
            }
        }
        #pragma unroll
        for (int c = 0; c < 8; ++c) {
            float bias = sB2[c * 16 + lo];
            #pragma unroll
            for (int v = 0; v < 8; ++v)
                hsc[(v + 8 * hi) * SH + c * 16 + lo] = fmaxf(acc[c][v] + bias, 0.0f);  // h2 over h1
        }
        asm volatile("s_wait_dscnt 0" ::: "memory");

        // ========== layer 3: Z = h2 @ W3 + b3 (single 16x16 tile) ==========
        v8f z;
        {
            v2f a = ldsv2(&hsc[lo * SH + 2 * hi]);
            v2f b = ldsv2(&sW3p[hi * PW3 + lo * 2]);
            #pragma unroll
            for (int kk = 0; kk < 32; ++kk) {
                v2f an = a, bn = b;
                if (kk < 31) {
                    an = ldsv2(&hsc[lo * SH + 4 * (kk + 1) + 2 * hi]);
                    bn = ldsv2(&sW3p[(2 * (kk + 1) + hi) * PW3 + lo * 2]);
                }
                z = __builtin_amdgcn_wmma_f32_16x16x4_f32(
                    false, a, false, b, (short)0, (kk == 0 ? zc : z), false, false);
                a = an; b = bn;
            }
        }

        // ========== V and X updates ==========
        // V += dt*0.5*sum_d Z^2 + sum_d Z * (sqdt*eps);  X = (1-dt)*X + sqdt*eps
        const float b3v = sB3[lo];
        #pragma unroll
        for (int v = 0; v < 8; ++v) {
            float zz = z[v] + b3v;
            float s  = 0.5f * dt * zz * zz + sqdt * zz * ep[v];
            s += __shfl_xor(s, 1);   // reduce over dims within each 16-lane half
            s += __shfl_xor(s, 2);
            s += __shfl_xor(s, 4);
            s += __shfl_xor(s, 8);
            va[v] += s;
            xr[v]  = decay * xr[v] + sqdt * ep[v];
        }
    }

    // ---------------- writeback ----------------
    #pragma unroll
    for (int v = 0; v < 8; ++v) {
        int row = rowbase + v + 8 * hi;
        outX[row * DD + lo] = xr[v];
        if (lo == 0) outV[row] = va[v];   // value replicated across each 16-lane half
    }
}

extern "C" void kernel_launch(void* const* d_in, const int* in_sizes, int n_in,
                              void* d_out, int out_size, void* d_ws, size_t ws_size,
                              hipStream_t stream) {
    (void)in_sizes; (void)n_in; (void)out_size; (void)d_ws; (void)ws_size;
    const float* X0    = (const float*)d_in[0];   // (N, D)
    const float* V0    = (const float*)d_in[1];   // (N,)
    const float* Yobs  = (const float*)d_in[2];   // (N, P)
    const float* noise = (const float*)d_in[3];   // (M, N, D)
    const float* W1    = (const float*)d_in[4];   // (25, 128)
    const float* b1    = (const float*)d_in[5];   // (128,)
    const float* W2    = (const float*)d_in[6];   // (128, 128)
    const float* b2    = (const float*)d_in[7];   // (128,)
    const float* W3    = (const float*)d_in[8];   // (128, 16)
    const float* b3    = (const float*)d_in[9];   // (16,)

    float* outX = (float*)d_out;          // N*D floats
    float* outV = outX + (size_t)NN * DD; // N floats

    dim3 grid(NN / (WAVES * 16));  // 256 blocks
    dim3 block(WAVES * 32);        // 128 threads = 4 waves
    sde_rollout_kernel<<<grid, block, 0, stream>>>(X0, V0, Yobs, noise,
                                                   W1, b1, W2, b2, W3, b3,
                                                   outX, outV);
}